// MambaMixer_59837484367983
// MI455X (gfx1250) — compile-verified
//
#include <hip/hip_runtime.h>
#include <hip/hip_bf16.h>

// ---------------- problem constants (from reference) ----------------
#define BATCH   2
#define LSEQ    1024
#define DMODEL  2048
#define DINNER  4096
#define DSTATE  16
#define DTRANK  128
#define KCONV   4
#define E2      (2 * DINNER)              // 8192 rows of xz
#define XDBL_E  (DTRANK + 2 * DSTATE)     // 160

// ---------------- GEMM tiling ----------------
#define BM    128        // block M footprint (8 waves: 2 wave-rows x 64)
#define BN    64         // block N footprint (4 wave-cols x 16)
#define KC    32         // K chunk staged in LDS
#define KSTR  36         // padded LDS K stride (even, 16B-friendly)

typedef float v2f __attribute__((ext_vector_type(2)));
typedef float v8f __attribute__((ext_vector_type(8)));

// =====================================================================
// fp32 WMMA GEMM:  C[m][n] = sum_k A[m][k] * B[k][n] (+ bias[m])
//   A row-major (lda).  B addressed B[k*b_rs + n*b_cs].
//   C addressed C[m*c_rs + n*c_cs].  *_bs = per-batch strides.
//   b_k_contig: 1 if B is contiguous along K (GEMM1), else row-major.
// Block: 256 thr = 8 waves; wave tile = 64(M) x 16(N) = 4 stacked WMMA
// tiles sharing one B fragment. Both A and B panels are staged in LDS;
// the next K-chunk is prefetched into registers while WMMAs run, so the
// inner loop is ds_load_b64 + v_wmma only (loadcnt waits sit at barriers).
// =====================================================================
__global__ __launch_bounds__(256)
void wmma_gemm_f32(const float* __restrict__ A,
                   const float* __restrict__ B,
                   float* __restrict__ C,
                   const float* __restrict__ bias,
                   int M, int N, int Kd,
                   long lda, long b_rs, long b_cs,
                   long c_rs, long c_cs,
                   long A_bs, long B_bs, long C_bs,
                   int b_k_contig)
{
    __shared__ float As[BM * KSTR];     // [m][k], padded k-stride
    __shared__ float Bs[BN * KSTR];     // [n][k], padded k-stride

    const int tid  = threadIdx.x;
    const int lane = tid & 31;
    const int wave = tid >> 5;           // 0..7
    const int tm   = wave >> 2;          // 0..1 : wave row (64 M each)
    const int tn   = wave & 3;           // 0..3 : wave col (16 N each)

    const long blockM = (long)blockIdx.y * BM;
    const long blockN = (long)blockIdx.x * BN;

    A += (long)blockIdx.z * A_bs;
    B += (long)blockIdx.z * B_bs;
    C += (long)blockIdx.z * C_bs;

    const int half = lane >> 4;          // 0: K pair {0,1}, 1: {2,3}
    const int l15  = lane & 15;
    const int kk   = half * 2;

    // ---- cooperative staging indices ----
    // A: 2 threads per row, 16 consecutive k each (4 x float4)
    const int a_row = tid >> 1;                       // 0..127
    const int a_kh  = (tid & 1) * 16;                 // 0 or 16
    long ar = blockM + a_row;
    const float* a_src = A + ((ar < (long)M) ? ar : 0) * lda + a_kh;
    // B row-major mode: 1 thread per (k, 8n); k-contig mode: 8 k per thread
    const int kloc = tid >> 3;                        // 0..31
    const int n8   = (tid & 7) * 8;                   // 8 consecutive n
    const int nloc = tid & 63;                        // 0..63
    const int kseg = (tid >> 6) * 8;                  // 0,8,16,24

    // ---- per-wave LDS read bases ----
    const float* bs_col = &Bs[(tn * 16 + l15) * KSTR];
    const float* as_row[4];
#pragma unroll
    for (int i = 0; i < 4; ++i)
        as_row[i] = &As[(tm * 64 + i * 16 + l15) * KSTR];

    v8f acc[4];
#pragma unroll
    for (int i = 0; i < 4; ++i)
        acc[i] = (v8f){0.f, 0.f, 0.f, 0.f, 0.f, 0.f, 0.f, 0.f};

    float4 aR[4];                        // staged A chunk (16 k per thread)
    float4 bR[2];                        // staged B chunk (8 elems per thread)

    auto load_chunk = [&](int kc) {
#pragma unroll
        for (int j = 0; j < 4; ++j)
            aR[j] = *(const float4*)(a_src + kc + 4 * j);
        if (b_k_contig) {
            const float* src = B + (blockN + nloc) * b_cs + kc + kseg;
            bR[0] = *(const float4*)(src);
            bR[1] = *(const float4*)(src + 4);
        } else {
            const float* src = B + (long)(kc + kloc) * b_rs + blockN + n8;
            bR[0] = *(const float4*)(src);
            bR[1] = *(const float4*)(src + 4);
        }
    };

    auto store_chunk = [&]() {
#pragma unroll
        for (int j = 0; j < 4; ++j)
            *(float4*)&As[a_row * KSTR + a_kh + 4 * j] = aR[j];
        if (b_k_contig) {
            *(float4*)&Bs[nloc * KSTR + kseg]     = bR[0];
            *(float4*)&Bs[nloc * KSTR + kseg + 4] = bR[1];
        } else {
            Bs[(n8 + 0) * KSTR + kloc] = bR[0].x;
            Bs[(n8 + 1) * KSTR + kloc] = bR[0].y;
            Bs[(n8 + 2) * KSTR + kloc] = bR[0].z;
            Bs[(n8 + 3) * KSTR + kloc] = bR[0].w;
            Bs[(n8 + 4) * KSTR + kloc] = bR[1].x;
            Bs[(n8 + 5) * KSTR + kloc] = bR[1].y;
            Bs[(n8 + 6) * KSTR + kloc] = bR[1].z;
            Bs[(n8 + 7) * KSTR + kloc] = bR[1].w;
        }
    };

    load_chunk(0);                       // prologue

    for (int kc = 0; kc < Kd; kc += KC) {
        store_chunk();
        __syncthreads();

        if (kc + KC < Kd)
            load_chunk(kc + KC);         // global loads overlap the WMMAs below

        // ---- compute: KC/4 = 8 WMMA steps, 4 M-tiles per step (LDS only) ----
#pragma unroll
        for (int k = 0; k < KC; k += 4) {
            v2f bfrag = *(const v2f*)(bs_col + k + kk);
            v2f a0 = *(const v2f*)(as_row[0] + k + kk);
            v2f a1 = *(const v2f*)(as_row[1] + k + kk);
            v2f a2 = *(const v2f*)(as_row[2] + k + kk);
            v2f a3 = *(const v2f*)(as_row[3] + k + kk);
            acc[0] = __builtin_amdgcn_wmma_f32_16x16x4_f32(false, a0, false, bfrag, (short)0, acc[0], false, false);
            acc[1] = __builtin_amdgcn_wmma_f32_16x16x4_f32(false, a1, false, bfrag, (short)0, acc[1], false, false);
            acc[2] = __builtin_amdgcn_wmma_f32_16x16x4_f32(false, a2, false, bfrag, (short)0, acc[2], false, false);
            acc[3] = __builtin_amdgcn_wmma_f32_16x16x4_f32(false, a3, false, bfrag, (short)0, acc[3], false, false);
        }
        __syncthreads();                 // WAR: LDS reads done before next store
    }

    // ---- store: VGPR e of tile i holds M = tileBase + e + 8*half ----
    const long ncol = blockN + tn * 16 + l15;
#pragma unroll
    for (int i = 0; i < 4; ++i) {
        const long mBase = blockM + tm * 64 + i * 16 + half * 8;
#pragma unroll
        for (int e = 0; e < 8; ++e) {
            const long m = mBase + e;
            if (m < (long)M) {
                float v = acc[i][e];
                if (bias) v += bias[m];
                C[m * c_rs + ncol * c_cs] = v;
            }
        }
    }
}

// =====================================================================
// Depthwise causal conv (K=4) + bias + SiLU; captures conv_state
// (raw x tail, pre-conv) as the reference does.
// =====================================================================
__global__ __launch_bounds__(256)
void conv_silu_kernel(const float* __restrict__ xz,     // [B][E2][L]
                      const float* __restrict__ conv_w, // [DINNER][K]
                      const float* __restrict__ conv_b, // [DINNER]
                      float* __restrict__ xc,           // [B][DINNER][L]
                      float* __restrict__ conv_state)   // [B][DINNER][K]
{
    const int l = blockIdx.x * blockDim.x + threadIdx.x;
    const int d = blockIdx.y;
    const int b = blockIdx.z;

    const float* xrow = xz + ((long)b * E2 + d) * LSEQ;   // x half: rows [0,DINNER)
    const float w0 = conv_w[d * KCONV + 0];
    const float w1 = conv_w[d * KCONV + 1];
    const float w2 = conv_w[d * KCONV + 2];
    const float w3 = conv_w[d * KCONV + 3];

    const float x3 = xrow[l];
    const float x2 = (l >= 1) ? xrow[l - 1] : 0.f;
    const float x1 = (l >= 2) ? xrow[l - 2] : 0.f;
    const float x0 = (l >= 3) ? xrow[l - 3] : 0.f;

    float acc = conv_b[d];
    acc = fmaf(w0, x0, acc);
    acc = fmaf(w1, x1, acc);
    acc = fmaf(w2, x2, acc);
    acc = fmaf(w3, x3, acc);

    const float s = acc / (1.f + __expf(-acc));           // SiLU
    xc[((long)b * DINNER + d) * LSEQ + l] = s;

    if (l >= LSEQ - KCONV)
        conv_state[((long)b * DINNER + d) * KCONV + (l - (LSEQ - KCONV))] = x3;
}

// =====================================================================
// Selective scan. One 16-lane group per (b,d) channel; lane = state idx n.
// Fuses softplus(dt + b_dt), A = -exp(A_log), +D*u skip, *silu(z) gate.
// y written into the (dead) x half of xz so GEMM4's B is row-major.
// =====================================================================
union F4 { float4 v; float f[4]; };

__global__ __launch_bounds__(256)
void scan_kernel(const float* __restrict__ dtraw,   // [B][DINNER][L]
                 const float* __restrict__ xc,      // [B][DINNER][L]
                 const float* __restrict__ xz,      // z half: rows [DINNER,E2)
                 const float* __restrict__ xdbl,    // [B][160][L]
                 const float* __restrict__ b_dt,    // [DINNER]
                 const float* __restrict__ A_log,   // [DINNER][16]
                 const float* __restrict__ Dp,      // [DINNER]
                 float* __restrict__ yout,          // alias of xz x-half
                 float* __restrict__ last_state)    // [B][DINNER][16]
{
    const int n   = threadIdx.x & 15;                 // state index
    const int grp = threadIdx.x >> 4;                 // 0..15 channel group in block
    const int ch  = blockIdx.x * 16 + grp;            // 0..B*DINNER-1
    const int b   = ch >> 12;                         // / DINNER
    const int d   = ch & (DINNER - 1);

    const float A_dn = -__expf(A_log[d * DSTATE + n]);
    const float Dd   = Dp[d];
    const float bdt  = b_dt[d];

    const float* drow = dtraw + ((long)b * DINNER + d) * LSEQ;
    const float* urow = xc    + ((long)b * DINNER + d) * LSEQ;
    const float* zrow = xz    + ((long)b * E2 + DINNER + d) * LSEQ;
    const float* Brow = xdbl  + ((long)b * XDBL_E + DTRANK + n) * LSEQ;
    const float* Crow = xdbl  + ((long)b * XDBL_E + DTRANK + DSTATE + n) * LSEQ;
    float*       yrow = yout  + ((long)b * E2 + d) * LSEQ;

    float st = 0.f;
    for (int t0 = 0; t0 < LSEQ; t0 += 4) {
        F4 dt4, u4, B4, C4, z4;
        dt4.v = *(const float4*)(drow + t0);
        u4.v  = *(const float4*)(urow + t0);
        B4.v  = *(const float4*)(Brow + t0);
        C4.v  = *(const float4*)(Crow + t0);
        z4.v  = *(const float4*)(zrow + t0);
#pragma unroll
        for (int j = 0; j < 4; ++j) {
            const float raw   = dt4.f[j] + bdt;
            const float delta = (raw > 20.f) ? raw : log1pf(__expf(raw)); // softplus
            const float u     = u4.f[j];
            const float dA    = __expf(delta * A_dn);
            st = fmaf(dA, st, delta * u * B4.f[j]);       // state update
            float yp = st * C4.f[j];
            // 16-lane butterfly sum (xor masks stay inside each 16-lane half)
            yp += __shfl_xor(yp, 1, 32);
            yp += __shfl_xor(yp, 2, 32);
            yp += __shfl_xor(yp, 4, 32);
            yp += __shfl_xor(yp, 8, 32);
            if (n == 0) {
                const float zv  = z4.f[j];
                const float sil = zv / (1.f + __expf(-zv));
                yrow[t0 + j] = (yp + Dd * u) * sil;
            }
        }
    }
    last_state[((long)b * DINNER + d) * DSTATE + n] = st;
}

// =====================================================================
// Host launch
// =====================================================================
extern "C" void kernel_launch(void* const* d_in, const int* in_sizes, int n_in,
                              void* d_out, int out_size, void* d_ws, size_t ws_size,
                              hipStream_t stream)
{
    const float* hidden = (const float*)d_in[0];
    const float* W_in   = (const float*)d_in[1];
    const float* conv_w = (const float*)d_in[2];
    const float* conv_b = (const float*)d_in[3];
    const float* W_x    = (const float*)d_in[4];
    const float* W_dt   = (const float*)d_in[5];
    const float* b_dt   = (const float*)d_in[6];
    const float* A_log  = (const float*)d_in[7];
    const float* Dp     = (const float*)d_in[8];
    const float* W_out  = (const float*)d_in[9];
    const float* b_out  = (const float*)d_in[10];

    float* out        = (float*)d_out;                        // [B][L][DMODEL]
    float* conv_state = out + (long)BATCH * LSEQ * DMODEL;    // [B][DINNER][K]
    float* last_state = conv_state + (long)BATCH * DINNER * KCONV;

    // workspace layout (floats): ~135.5 MB total
    float* ws    = (float*)d_ws;
    float* xz    = ws;                                   // [B][E2][L]      67 MB
    float* xc    = xz   + (long)BATCH * E2 * LSEQ;       // [B][DINNER][L]  33.5 MB
    float* xdbl  = xc   + (long)BATCH * DINNER * LSEQ;   // [B][160][L]      1.3 MB
    float* dtraw = xdbl + (long)BATCH * XDBL_E * LSEQ;   // [B][DINNER][L]  33.5 MB

    const dim3 blk(256);

    // GEMM1: xz[b][e][l] = sum_d W_in[e][d] * hidden[b][l][d]
    wmma_gemm_f32<<<dim3(LSEQ / BN, (E2 + BM - 1) / BM, BATCH), blk, 0, stream>>>(
        W_in, hidden, xz, nullptr,
        E2, LSEQ, DMODEL,
        (long)DMODEL,                 // lda
        1L, (long)DMODEL,             // B[k][n] = hidden[n*DMODEL + k] (K contig)
        (long)LSEQ, 1L,               // C row-major (e,l)
        0L, (long)LSEQ * DMODEL, (long)E2 * LSEQ,
        /*b_k_contig=*/1);

    // depthwise conv + SiLU; capture conv_state
    conv_silu_kernel<<<dim3(LSEQ / 256, DINNER, BATCH), blk, 0, stream>>>(
        xz, conv_w, conv_b, xc, conv_state);

    // GEMM2: xdbl[b][e][l] = sum_d W_x[e][d] * xc[b][d][l]   (stored (e,l))
    wmma_gemm_f32<<<dim3(LSEQ / BN, (XDBL_E + BM - 1) / BM, BATCH), blk, 0, stream>>>(
        W_x, xc, xdbl, nullptr,
        XDBL_E, LSEQ, DINNER,
        (long)DINNER,
        (long)LSEQ, 1L,               // B row-major (d,l)
        (long)LSEQ, 1L,
        0L, (long)DINNER * LSEQ, (long)XDBL_E * LSEQ,
        /*b_k_contig=*/0);

    // GEMM3: dtraw[b][d][l] = sum_r W_dt[d][r] * xdbl[b][r][l]
    wmma_gemm_f32<<<dim3(LSEQ / BN, (DINNER + BM - 1) / BM, BATCH), blk, 0, stream>>>(
        W_dt, xdbl, dtraw, nullptr,
        DINNER, LSEQ, DTRANK,
        (long)DTRANK,
        (long)LSEQ, 1L,
        (long)LSEQ, 1L,
        0L, (long)XDBL_E * LSEQ, (long)DINNER * LSEQ,
        /*b_k_contig=*/0);

    // selective scan (writes y into x-half of xz; emits last_state)
    scan_kernel<<<dim3((BATCH * DINNER) / 16), blk, 0, stream>>>(
        dtraw, xc, xz, xdbl, b_dt, A_log, Dp, xz, last_state);

    // GEMM4: out[b][l][e] = sum_d W_out[e][d] * y[b][d][l] + b_out[e]
    wmma_gemm_f32<<<dim3(LSEQ / BN, (DMODEL + BM - 1) / BM, BATCH), blk, 0, stream>>>(
        W_out, xz, out, b_out,
        DMODEL, LSEQ, DINNER,
        (long)DINNER,
        (long)LSEQ, 1L,               // y row-major (d,l)
        1L, (long)DMODEL,             // store transposed into [l][e]
        0L, (long)E2 * LSEQ, (long)LSEQ * DMODEL,
        /*b_k_contig=*/0);
}